// HeterogeneousGNNEncoder_60610578481528
// MI455X (gfx1250) — compile-verified
//
#include <hip/hip_runtime.h>
#include <math.h>

#define HID 128
#define NHEAD 4
#define HDIM 32
#define NLAYERS 2
#define NBATCH 4
#define NEG_SLOPE 0.2f
#define LN_EPS 1e-5f
#define APAD 132   // LDS row stride (floats): 132 % 64 = 4 -> conflict-free column reads

typedef __attribute__((ext_vector_type(2))) float v2f;
typedef __attribute__((ext_vector_type(8))) float v8f;

static inline int ceildiv(int a, int b) { return (a + b - 1) / b; }

// ---------------- device helpers ----------------

__device__ inline void atomicMaxF(float* addr, float val) {
  unsigned int* ua = (unsigned int*)addr;
  unsigned int old = *ua;
  while (__uint_as_float(old) < val) {
    unsigned int assumed = old;
    old = atomicCAS(ua, assumed, __float_as_uint(val));
    if (old == assumed) break;
  }
}

// async 16B global -> LDS copy (GLOBAL_LOAD_ASYNC_TO_LDS_B128, ASYNCcnt-tracked).
// LDS operand is the wave-relative offset = low 32 bits of the generic pointer
// (flat aperture mapping).
__device__ inline void async_g2l_b128(const float* gsrc, float* ldst) {
  unsigned int laddr = (unsigned int)(unsigned long long)ldst;
  asm volatile("global_load_async_to_lds_b128 %0, %1, off"
               :: "v"(laddr), "v"(gsrc)
               : "memory");
}

__device__ inline void wait_async_all() {
  asm volatile("s_wait_asynccnt 0" ::: "memory");
}

// ---------------- kernels ----------------

// Y[N,128] = X[N,D] @ W[D,128] + b   (small D: 3..10, plain VALU)
__global__ void proj_kernel(const float* __restrict__ X, const float* __restrict__ W,
                            const float* __restrict__ b, float* __restrict__ Y,
                            int N, int D) {
  int idx = blockIdx.x * blockDim.x + threadIdx.x;
  if (idx >= N * HID) return;
  int n = idx / HID, f = idx % HID;
  float acc = b[f];
  for (int k = 0; k < D; ++k) acc += X[n * D + k] * W[k * HID + f];
  Y[idx] = acc;
}

// H[N,128] = X[N,128] @ W[128,128]  -- f32 WMMA 16x16x4.
// The 16x128 A stripe is staged once per block into LDS via async copies
// (8 waves share it instead of re-reading global 8x); each wave owns a
// 16-column tile of B/D.
__global__ __launch_bounds__(256)
void gemm128_wmma(const float* __restrict__ X, const float* __restrict__ W,
                  float* __restrict__ H, int N) {
  __shared__ float shA[16 * APAD];
  const int tid = threadIdx.x;
  const int rowBase = blockIdx.x * 16;

  // stage A stripe: 512 float4 chunks, 2 per thread
#pragma unroll
  for (int v = 0; v < 2; ++v) {
    int e4 = tid + v * 256;          // float4 index: 16 rows x 32 chunks
    int r = e4 >> 5;
    int c4 = e4 & 31;
    int gr = rowBase + r;
    if (gr >= N) gr = N - 1;         // clamp (EXEC must stay full for WMMA anyway)
    async_g2l_b128(X + (size_t)gr * HID + c4 * 4, &shA[r * APAD + c4 * 4]);
  }
  wait_async_all();
  __syncthreads();

  const int lane = threadIdx.x & 31;
  const int wave = threadIdx.x >> 5;       // 0..7 -> column tile
  const int lo = lane & 15;
  const int hi = lane >> 4;                // 0/1: which half of the wave
  const int col = wave * 16 + lo;
  const float* ar = &shA[lo * APAD];
  v8f acc = {};
#pragma unroll
  for (int kb = 0; kb < HID; kb += 4) {
    v2f a, b;
    // A 16x4 f32: VGPR j, half hi -> K = j + 2*hi (ISA 7.12.2)
    a.x = ar[kb + 2 * hi + 0];
    a.y = ar[kb + 2 * hi + 1];
    // B 4x16 f32: same K mapping, N = lane%16
    b.x = W[(size_t)(kb + 2 * hi + 0) * HID + col];
    b.y = W[(size_t)(kb + 2 * hi + 1) * HID + col];
    acc = __builtin_amdgcn_wmma_f32_16x16x4_f32(false, a, false, b, (short)0, acc,
                                                false, false);
  }
  // D 16x16 f32: VGPR j -> row j + 8*hi
  const int rbase = rowBase + 8 * hi;
#pragma unroll
  for (int j = 0; j < 8; ++j) {
    int r = rbase + j;
    if (r < N) H[(size_t)r * HID + col] = acc[j];
  }
}

// AS[n,h] = sum_d HS[n, h*32+d] * a_src[h,d]
__global__ void as_kernel(const float* __restrict__ HS, const float* __restrict__ a_src,
                          float* __restrict__ AS, int N) {
  int idx = blockIdx.x * blockDim.x + threadIdx.x;
  if (idx >= N * NHEAD) return;
  int n = idx >> 2, h = idx & 3;
  const float* hp = HS + (size_t)n * HID + h * HDIM;
  const float* ap = a_src + h * HDIM;
  float acc = 0.f;
#pragma unroll
  for (int d = 0; d < HDIM; ++d) acc += hp[d] * ap[d];
  AS[idx] = acc;
}

// W2[k,h] = sum_d W[k, h*32+d] * a_dst[h,d]   (folds hd . a_dst)
__global__ void w2_kernel(const float* __restrict__ W, const float* __restrict__ a_dst,
                          float* __restrict__ W2) {
  int idx = blockIdx.x * blockDim.x + threadIdx.x;
  if (idx >= HID * NHEAD) return;
  int k = idx >> 2, h = idx & 3;
  float acc = 0.f;
#pragma unroll
  for (int d = 0; d < HDIM; ++d) acc += W[(size_t)k * HID + h * HDIM + d] * a_dst[h * HDIM + d];
  W2[idx] = acc;
}

// AD[n,h] = sum_k X[n,k] * W2[k,h]
__global__ void ad_kernel(const float* __restrict__ X, const float* __restrict__ W2,
                          float* __restrict__ AD, int N) {
  int idx = blockIdx.x * blockDim.x + threadIdx.x;
  if (idx >= N * NHEAD) return;
  int n = idx >> 2, h = idx & 3;
  const float* xp = X + (size_t)n * HID;
  float acc = 0.f;
  for (int k = 0; k < HID; ++k) acc += xp[k] * W2[k * NHEAD + h];
  AD[idx] = acc;
}

__global__ void fill_kernel(float* __restrict__ p, float v, int n) {
  int idx = blockIdx.x * blockDim.x + threadIdx.x;
  if (idx < n) p[idx] = v;
}

// SUMS[n,f] = sum of biases of incoming edge types (up to 3)
__global__ void bias3_kernel(float* __restrict__ S, const float* __restrict__ b0,
                             const float* __restrict__ b1, const float* __restrict__ b2,
                             int N) {
  int idx = blockIdx.x * blockDim.x + threadIdx.x;
  if (idx >= N * HID) return;
  int f = idx % HID;
  float v = 0.f;
  if (b0) v += b0[f];
  if (b1) v += b1[f];
  if (b2) v += b2[f];
  S[idx] = v;
}

__device__ inline float lrelu(float a) { return a > 0.f ? a : NEG_SLOPE * a; }

__global__ void edge_max_kernel(const int* __restrict__ src, const int* __restrict__ dst,
                                const float* __restrict__ AS, const float* __restrict__ AD,
                                float* __restrict__ M, int E) {
  int e = blockIdx.x * blockDim.x + threadIdx.x;
  if (e >= E) return;
  int s = src[e], d = dst[e];
#pragma unroll
  for (int h = 0; h < NHEAD; ++h) {
    float a = lrelu(AS[s * NHEAD + h] + AD[d * NHEAD + h]);
    atomicMaxF(&M[d * NHEAD + h], a);
  }
}

__global__ void fixm_kernel(float* __restrict__ M, int n) {
  int idx = blockIdx.x * blockDim.x + threadIdx.x;
  if (idx >= n) return;
  float v = M[idx];
  if (v == -INFINITY) M[idx] = 0.f;   // guard empty segments
}

__global__ void edge_sum_kernel(const int* __restrict__ src, const int* __restrict__ dst,
                                const float* __restrict__ AS, const float* __restrict__ AD,
                                const float* __restrict__ M, float* __restrict__ DEN, int E) {
  int e = blockIdx.x * blockDim.x + threadIdx.x;
  if (e >= E) return;
  int s = src[e], d = dst[e];
#pragma unroll
  for (int h = 0; h < NHEAD; ++h) {
    float a = lrelu(AS[s * NHEAD + h] + AD[d * NHEAD + h]);
    atomicAdd(&DEN[d * NHEAD + h], expf(a - M[d * NHEAD + h]));
  }
}

// one wave32 per edge: SUMS[dst,f] += HS[src,f] * alpha[f/32]
__global__ __launch_bounds__(256)
void edge_scatter_kernel(const int* __restrict__ src, const int* __restrict__ dst,
                         const float* __restrict__ AS, const float* __restrict__ AD,
                         const float* __restrict__ M, const float* __restrict__ DEN,
                         const float* __restrict__ HS, float* __restrict__ SUMS, int E) {
  int e = blockIdx.x * 8 + (threadIdx.x >> 5);
  if (e >= E) return;
  int lane = threadIdx.x & 31;
  int s = src[e], d = dst[e];
#pragma unroll
  for (int h = 0; h < NHEAD; ++h) {
    float a = lrelu(AS[s * NHEAD + h] + AD[d * NHEAD + h]);
    float alpha = expf(a - M[d * NHEAD + h]) / (DEN[d * NHEAD + h] + 1e-16f);
    float v = HS[(size_t)s * HID + h * HDIM + lane] * alpha;
    atomicAdd(&SUMS[(size_t)d * HID + h * HDIM + lane], v);
  }
}

// x = LN(relu(sums*invCnt) + x) * g + b   -- one wave32 per node, in place
__global__ __launch_bounds__(256)
void epilogue_kernel(const float* __restrict__ SUMS, float* __restrict__ X,
                     const float* __restrict__ g, const float* __restrict__ bln,
                     float invCnt, int N) {
  int n = blockIdx.x * 8 + (threadIdx.x >> 5);
  if (n >= N) return;
  int lane = threadIdx.x & 31;
  float y[4];
  float part = 0.f;
#pragma unroll
  for (int j = 0; j < 4; ++j) {
    int f = j * 32 + lane;
    float sv = SUMS[(size_t)n * HID + f] * invCnt;
    sv = sv > 0.f ? sv : 0.f;
    y[j] = sv + X[(size_t)n * HID + f];
    part += y[j];
  }
#pragma unroll
  for (int off = 16; off > 0; off >>= 1) part += __shfl_xor(part, off, 32);
  float mu = part * (1.0f / HID);
  float vp = 0.f;
#pragma unroll
  for (int j = 0; j < 4; ++j) { float t = y[j] - mu; vp += t * t; }
#pragma unroll
  for (int off = 16; off > 0; off >>= 1) vp += __shfl_xor(vp, off, 32);
  float rs = rsqrtf(vp * (1.0f / HID) + LN_EPS);
#pragma unroll
  for (int j = 0; j < 4; ++j) {
    int f = j * 32 + lane;
    X[(size_t)n * HID + f] = (y[j] - mu) * rs * g[f] + bln[f];
  }
}

__global__ void pool_acc_kernel(const float* __restrict__ X, const int* __restrict__ batch,
                                float* __restrict__ pool, float* __restrict__ pcnt, int N) {
  int idx = blockIdx.x * blockDim.x + threadIdx.x;
  if (idx >= N * HID) return;
  int n = idx / HID, f = idx % HID;
  int b = batch[n];
  atomicAdd(&pool[(size_t)b * HID + f], X[idx]);
  if (f == 0) atomicAdd(&pcnt[b], 1.0f);
}

// out[b, t*128+f] = pool[t][b][f] / max(cnt,1)
__global__ void pool_fin_kernel(const float* __restrict__ POOL, const float* __restrict__ PCNT,
                                float* __restrict__ out) {
  int idx = blockIdx.x * blockDim.x + threadIdx.x;
  if (idx >= NBATCH * 4 * HID) return;
  int b = idx / (4 * HID);
  int rem = idx % (4 * HID);
  int t = rem / HID, f = rem % HID;
  float c = PCNT[t * NBATCH + b];
  c = c > 1.0f ? c : 1.0f;
  out[idx] = POOL[((size_t)t * NBATCH + b) * HID + f] / c;
}

// ---------------- host orchestration ----------------

extern "C" void kernel_launch(void* const* d_in, const int* in_sizes, int n_in,
                              void* d_out, int out_size, void* d_ws, size_t ws_size,
                              hipStream_t stream) {
  (void)n_in; (void)out_size; (void)ws_size;
  const int NN[4] = {100000, 2000, 1000, 200};           // block, spmt, crane, facility
  const int NDm[4] = {8, 10, 7, 3};
  const int ESRC[8] = {0, 1, 0, 2, 0, 0, 1, 2};
  const int EDSTt[8] = {1, 0, 2, 0, 3, 0, 3, 3};
  const int CNT[4] = {3, 1, 1, 3};                       // incoming edge-type counts
  const int INC[4][3] = {{1, 3, 5}, {0, -1, -1}, {2, -1, -1}, {4, 6, 7}};

  // inputs: 0-3 x_*, 4-19 src_i/dst_i, 20-23 batch_*,
  // params (insertion order): 24-31 proj (W,b)x4,
  //   32 + (l*8+i)*4 + {0:W,1:b,2:a_src,3:a_dst}, 96 + 2*l {ln.g, ln.b}
  const float* xin[4]; const int* srcp[8]; const int* dstp[8]; const int* batchp[4];
  int EC[8];
  for (int t = 0; t < 4; ++t) xin[t] = (const float*)d_in[t];
  for (int i = 0; i < 8; ++i) {
    srcp[i] = (const int*)d_in[4 + 2 * i];
    dstp[i] = (const int*)d_in[5 + 2 * i];
    EC[i] = in_sizes[4 + 2 * i];
  }
  for (int t = 0; t < 4; ++t) batchp[t] = (const int*)d_in[20 + t];

  float* ws = (float*)d_ws;
  size_t off = 0;
  size_t offX[4], offS[4];
  for (int t = 0; t < 4; ++t) { offX[t] = off; off += (size_t)NN[t] * HID; }
  for (int t = 0; t < 4; ++t) { offS[t] = off; off += (size_t)NN[t] * HID; }
  float* HS = ws + off;  off += (size_t)100000 * HID;
  float* AS = ws + off;  off += 400000;
  float* AD = ws + off;  off += 400000;
  float* Mx = ws + off;  off += 400000;
  float* DEN = ws + off; off += 400000;
  float* W2 = ws + off;  off += HID * NHEAD;
  float* POOL = ws + off; off += (size_t)4 * NBATCH * HID;
  float* PCNT = ws + off; off += 4 * NBATCH;

  // initial per-type projection
  for (int t = 0; t < 4; ++t) {
    const float* Wp = (const float*)d_in[24 + 2 * t];
    const float* bp = (const float*)d_in[25 + 2 * t];
    proj_kernel<<<ceildiv(NN[t] * HID, 256), 256, 0, stream>>>(
        xin[t], Wp, bp, ws + offX[t], NN[t], NDm[t]);
  }

  for (int l = 0; l < NLAYERS; ++l) {
    // init per-dest accumulators with summed incoming biases
    for (int t = 0; t < 4; ++t) {
      const float* bb[3] = {nullptr, nullptr, nullptr};
      for (int j = 0; j < 3; ++j)
        if (INC[t][j] >= 0) bb[j] = (const float*)d_in[32 + (l * 8 + INC[t][j]) * 4 + 1];
      bias3_kernel<<<ceildiv(NN[t] * HID, 256), 256, 0, stream>>>(
          ws + offS[t], bb[0], bb[1], bb[2], NN[t]);
    }
    for (int i = 0; i < 8; ++i) {
      int s = ESRC[i], d = EDSTt[i];
      int base = 32 + (l * 8 + i) * 4;
      const float* W = (const float*)d_in[base + 0];
      const float* a_src = (const float*)d_in[base + 2];
      const float* a_dst = (const float*)d_in[base + 3];

      gemm128_wmma<<<ceildiv(NN[s], 16), 256, 0, stream>>>(ws + offX[s], W, HS, NN[s]);
      as_kernel<<<ceildiv(NN[s] * NHEAD, 256), 256, 0, stream>>>(HS, a_src, AS, NN[s]);
      w2_kernel<<<ceildiv(HID * NHEAD, 256), 256, 0, stream>>>(W, a_dst, W2);
      ad_kernel<<<ceildiv(NN[d] * NHEAD, 256), 256, 0, stream>>>(ws + offX[d], W2, AD, NN[d]);

      fill_kernel<<<ceildiv(NN[d] * NHEAD, 256), 256, 0, stream>>>(Mx, -INFINITY, NN[d] * NHEAD);
      fill_kernel<<<ceildiv(NN[d] * NHEAD, 256), 256, 0, stream>>>(DEN, 0.0f, NN[d] * NHEAD);

      edge_max_kernel<<<ceildiv(EC[i], 256), 256, 0, stream>>>(srcp[i], dstp[i], AS, AD, Mx, EC[i]);
      fixm_kernel<<<ceildiv(NN[d] * NHEAD, 256), 256, 0, stream>>>(Mx, NN[d] * NHEAD);
      edge_sum_kernel<<<ceildiv(EC[i], 256), 256, 0, stream>>>(srcp[i], dstp[i], AS, AD, Mx, DEN, EC[i]);
      edge_scatter_kernel<<<ceildiv(EC[i], 8), 256, 0, stream>>>(
          srcp[i], dstp[i], AS, AD, Mx, DEN, HS, ws + offS[d], EC[i]);
    }
    const float* g = (const float*)d_in[96 + 2 * l];
    const float* bl = (const float*)d_in[97 + 2 * l];
    for (int t = 0; t < 4; ++t)
      epilogue_kernel<<<ceildiv(NN[t], 8), 256, 0, stream>>>(
          ws + offS[t], ws + offX[t], g, bl, 1.0f / CNT[t], NN[t]);
  }

  // global mean pool per node type, concat
  fill_kernel<<<ceildiv(4 * NBATCH * HID + 4 * NBATCH, 256), 256, 0, stream>>>(
      POOL, 0.0f, 4 * NBATCH * HID + 4 * NBATCH);
  for (int t = 0; t < 4; ++t)
    pool_acc_kernel<<<ceildiv(NN[t] * HID, 256), 256, 0, stream>>>(
        ws + offX[t], batchp[t], POOL + (size_t)t * NBATCH * HID, PCNT + t * NBATCH, NN[t]);
  pool_fin_kernel<<<ceildiv(NBATCH * 4 * HID, 256), 256, 0, stream>>>(
      POOL, PCNT, (float*)d_out);
}